// DAGNN_10900626997641
// MI455X (gfx1250) — compile-verified
//
#include <hip/hip_runtime.h>
#include <hip/hip_bf16.h>

// DAGNN: out = sum_k sigmoid(H_k . s) * H_k, H_0 = (relu(x@W1+b1))@W2+b2,
//        H_{k+1} = Anorm @ H_k  (10 hops, 3.3M-edge COO SpMM).
// Attention score for hop k depends only on hop k -> stream-accumulate into
// out, never materializing the [N,11,64] stack. MLP is fused through LDS so
// the [N,512] hidden tensor is never written to HBM.

typedef __attribute__((ext_vector_type(2))) float v2f;
typedef __attribute__((ext_vector_type(4))) float v4f;
typedef __attribute__((ext_vector_type(8))) float v8f;

#define IN_DIM 512
#define HID_DIM 512
#define OUT_DIM 64
#define LDS_PITCH 68  // 16x64 fp32 tile, padded pitch -> conflict-free LDS

// ---------------------------------------------------------------------------
// Tiled transpose: out[c*R + r] = in[r*C + c].  Run once per call on W1/W2 so
// the WMMA B-operand pair (two consecutive K rows) becomes one contiguous
// global_load_b64 instead of two strided b32 loads.
// ---------------------------------------------------------------------------
__global__ __launch_bounds__(256) void k_transpose(
    const float* __restrict__ in, float* __restrict__ out, int R, int C)
{
    __shared__ float t[32][33];
    const int tx = threadIdx.x, ty = threadIdx.y;
    const int c0 = blockIdx.x * 32, r0 = blockIdx.y * 32;
    for (int i = 0; i < 4; ++i) {
        int r = r0 + ty + 8 * i, c = c0 + tx;
        if (r < R && c < C) t[ty + 8 * i][tx] = in[(size_t)r * C + c];
    }
    __syncthreads();
    for (int i = 0; i < 4; ++i) {
        int c = c0 + ty + 8 * i, r = r0 + tx;
        if (r < R && c < C) out[(size_t)c * R + r] = t[tx][ty + 8 * i];
    }
}

// ---------------------------------------------------------------------------
// Fused MLP: h0[m, 0:64] = relu(x[m,:] @ W1 + b1) @ W2 + b2
// Block = 128 threads (4 waves); each wave owns a 16x64 output tile built
// from V_WMMA_F32_16X16X4_F32. Hidden activations produced 64 cols at a
// time, ReLU'd, re-laid-out D->A through padded LDS, folded into GEMM2.
// Wt1[n][k] = W1[k][n], Wt2[n][k] = W2[k][n] (pre-transposed).
// ---------------------------------------------------------------------------
__global__ __launch_bounds__(128) void k_mlp(
    const float* __restrict__ x, const float* __restrict__ Wt1,
    const float* __restrict__ b1, const float* __restrict__ Wt2,
    const float* __restrict__ b2, float* __restrict__ h0, int nrows)
{
    __shared__ float lds_h[4 * 16 * LDS_PITCH];

    const int wave = threadIdx.x >> 5;
    const int lane = threadIdx.x & 31;
    const int lm   = lane & 15;   // M (A) / N (B,C,D) within tile
    const int hi   = lane >> 4;   // K-pair select (A,B) / row-half (C,D)

    const int m0 = (blockIdx.x * 4 + wave) * 16;
    if (m0 >= nrows) return;  // wave-uniform; EXEC all-ones for WMMA

    float* lds = &lds_h[wave * 16 * LDS_PITCH];

    v8f oacc[4];
    for (int j = 0; j < 4; ++j) {
        const float bv = b2[16 * j + lm];
        for (int v = 0; v < 8; ++v) oacc[j][v] = bv;
    }

    const float* xrow = x + (size_t)(m0 + lm) * IN_DIM;

    for (int c = 0; c < HID_DIM / 64; ++c) {
        const int n0 = c * 64;

        // ---- GEMM1 chunk: hacc[16x64] = x[16x512] @ W1[:, n0:n0+64] + b1 ----
        v8f hacc[4];
        for (int j = 0; j < 4; ++j) {
            const float bv = b1[n0 + 16 * j + lm];
            for (int v = 0; v < 8; ++v) hacc[j][v] = bv;
        }
        for (int k = 0; k < IN_DIM; k += 4) {
            // A 16x4: lanes 0-15 hold K=k,k+1; lanes 16-31 hold K=k+2,k+3
            v2f a = *(const v2f*)(xrow + k + 2 * hi);
            for (int j = 0; j < 4; ++j) {
                // B 4x16: Wt1[n][k+2hi .. k+2hi+1] -> single b64 load
                v2f b = *(const v2f*)(Wt1 + (size_t)(n0 + 16 * j + lm) * IN_DIM
                                      + k + 2 * hi);
                hacc[j] = __builtin_amdgcn_wmma_f32_16x16x4_f32(
                    false, a, false, b, (short)0, hacc[j], false, false);
            }
        }

        // ---- ReLU + D-layout -> LDS (pitch 68 -> conflict-free) ----
        for (int j = 0; j < 4; ++j)
            for (int v = 0; v < 8; ++v) {
                float val = hacc[j][v];
                val = val > 0.0f ? val : 0.0f;
                lds[(v + 8 * hi) * LDS_PITCH + 16 * j + lm] = val;
            }
        // Per-wave private LDS region; same-wave DS ops stay in order.

        // ---- GEMM2 chunk: oacc += hchunk[16x64] @ W2[n0:n0+64, 0:64] ----
        for (int kk = 0; kk < 64; kk += 4) {
            v2f a2 = *(const v2f*)(lds + lm * LDS_PITCH + kk + 2 * hi);
            for (int j = 0; j < 4; ++j) {
                v2f b = *(const v2f*)(Wt2 + (size_t)(16 * j + lm) * HID_DIM
                                      + n0 + kk + 2 * hi);
                oacc[j] = __builtin_amdgcn_wmma_f32_16x16x4_f32(
                    false, a2, false, b, (short)0, oacc[j], false, false);
            }
        }
    }

    // ---- Store 16x64 tile: D VGPR v holds row (v + 8*hi), col 16*j+lm ----
    for (int j = 0; j < 4; ++j)
        for (int v = 0; v < 8; ++v)
            h0[(size_t)(m0 + v + 8 * hi) * OUT_DIM + 16 * j + lm] = oacc[j][v];
}

// ---------------------------------------------------------------------------
// SpMM: hout[row[e], :] += w[e] * hin[col[e], :].  16 lanes per edge, each
// lane moves a float4 -> 256B contiguous b128 gather per edge; scatter via
// hardware f32 atomics (global_atomic_add_f32, no CAS loop).
// ---------------------------------------------------------------------------
__global__ __launch_bounds__(256) void k_spmm(
    const int* __restrict__ row, const int* __restrict__ col,
    const float* __restrict__ w, const float* __restrict__ hin,
    float* __restrict__ hout, int nedges)
{
    const int t = blockIdx.x * 256 + threadIdx.x;
    const int e = t >> 4;
    const int q = t & 15;
    if (e >= nedges) return;

    const int r  = row[e];
    const int cI = col[e];
    const float ww = w[e];

    v4f hv = *(const v4f*)(hin + (size_t)cI * OUT_DIM + 4 * q);
    float* dst = hout + (size_t)r * OUT_DIM + 4 * q;
    unsafeAtomicAdd(dst + 0, ww * hv.x);
    unsafeAtomicAdd(dst + 1, ww * hv.y);
    unsafeAtomicAdd(dst + 2, ww * hv.z);
    unsafeAtomicAdd(dst + 3, ww * hv.w);
}

// ---------------------------------------------------------------------------
// Combine: per node n, z = h[n,:]·s ; g = sigmoid(z) ; out[n,:] (+)= g*h[n,:]
// 16 lanes per node (two nodes per wave), float4 per lane, 4-step shfl_xor
// reduction (masks 8..1 stay inside each 16-lane half).
// ---------------------------------------------------------------------------
__global__ __launch_bounds__(256) void k_combine(
    const float* __restrict__ h, const float* __restrict__ s,
    float* __restrict__ out, int nnodes, int beta)
{
    const int t = blockIdx.x * 256 + threadIdx.x;
    const int n = t >> 4;
    const int q = t & 15;
    if (n >= nnodes) return;

    v4f hv = *(const v4f*)(h + (size_t)n * OUT_DIM + 4 * q);
    v4f sv = *(const v4f*)(s + 4 * q);
    float p = hv.x * sv.x + hv.y * sv.y + hv.z * sv.z + hv.w * sv.w;
    for (int off = 8; off >= 1; off >>= 1)
        p += __shfl_xor(p, off, 32);
    const float g = 1.0f / (1.0f + __expf(-p));

    float* op = out + (size_t)n * OUT_DIM + 4 * q;
    v4f o;
    o.x = g * hv.x; o.y = g * hv.y; o.z = g * hv.z; o.w = g * hv.w;
    if (beta) { o.x += op[0]; o.y += op[1]; o.z += op[2]; o.w += op[3]; }
    *(v4f*)op = o;
}

// ---------------------------------------------------------------------------
extern "C" void kernel_launch(void* const* d_in, const int* in_sizes, int n_in,
                              void* d_out, int out_size, void* d_ws, size_t ws_size,
                              hipStream_t stream) {
    const float* x   = (const float*)d_in[0];
    const int*   row = (const int*)d_in[1];
    const int*   col = (const int*)d_in[2];
    const float* ew  = (const float*)d_in[3];
    const float* W1  = (const float*)d_in[4];
    const float* b1  = (const float*)d_in[5];
    const float* W2  = (const float*)d_in[6];
    const float* b2  = (const float*)d_in[7];
    const float* s   = (const float*)d_in[8];
    float* out = (float*)d_out;

    const int N = in_sizes[0] / IN_DIM;   // 100000
    const int E = in_sizes[1];            // 3300000

    // Workspace: hA/hB ping-pong [N,64] + transposed weights (~52.3 MB).
    float* hA  = (float*)d_ws;
    float* hB  = hA + (size_t)N * OUT_DIM;
    float* Wt1 = hB + (size_t)N * OUT_DIM;                 // [HID][IN]
    float* Wt2 = Wt1 + (size_t)IN_DIM * HID_DIM;           // [OUT][HID]

    // One-time per call: transpose weights for b64 B-operand loads.
    k_transpose<<<dim3(HID_DIM / 32, IN_DIM / 32), dim3(32, 8), 0, stream>>>(
        W1, Wt1, IN_DIM, HID_DIM);
    k_transpose<<<dim3(OUT_DIM / 32, HID_DIM / 32), dim3(32, 8), 0, stream>>>(
        W2, Wt2, HID_DIM, OUT_DIM);

    // H_0 = (relu(x@W1+b1)) @ W2 + b2
    k_mlp<<<(N + 63) / 64, 128, 0, stream>>>(x, Wt1, b1, Wt2, b2, hA, N);
    // out = sigmoid(H_0·s) * H_0   (beta=0: overwrite poisoned d_out)
    k_combine<<<(N * 16 + 255) / 256, 256, 0, stream>>>(hA, s, out, N, 0);

    float* cur = hA;
    float* nxt = hB;
    for (int hop = 0; hop < 10; ++hop) {
        hipMemsetAsync(nxt, 0, (size_t)N * OUT_DIM * sizeof(float), stream);
        k_spmm<<<((size_t)E * 16 + 255) / 256, 256, 0, stream>>>(
            row, col, ew, cur, nxt, E);
        k_combine<<<(N * 16 + 255) / 256, 256, 0, stream>>>(nxt, s, out, N, 1);
        float* tmp = cur; cur = nxt; nxt = tmp;
    }
}